// Worker_70746701300061
// MI455X (gfx1250) — compile-verified
//
#include <hip/hip_runtime.h>

// CDNA5 / gfx1250 fused kernel: edge-embed GEMM + gather + normalize + 2-layer
// MLP + head, all on V_WMMA_F32_16X16X4_F32 (full f32 precision, wave32).
//
// Layout convention (per wave, 16 pairs per tile):
//   A operand (weights):  lane holds W[mt*16 + lane%16][kb + 2*(lane/16) + {0,1}]
//   B operand (acts):     lane holds X[pair = lane%16][kb + 2*(lane/16) + {0,1}]
//   C/D:                  reg r, lane l -> (ch = mt*16 + r + 8*(l>>4), pair = l&15)

typedef __attribute__((ext_vector_type(2))) float v2f;
typedef __attribute__((ext_vector_type(8))) float v8f;

#define NPAIRS 500000
#define FEAT   256
#define EMB    64
#define U1DIM  128
#define WPB    4            // waves per block
#define THREADS (WPB * 32)

static __device__ __forceinline__ v8f wmma4(v2f a, v2f b, v8f c) {
  // 8 args: (neg_a, A, neg_b, B, c_mod, C, reuse_a, reuse_b)
  return __builtin_amdgcn_wmma_f32_16x16x4_f32(false, a, false, b, (short)0, c,
                                               false, false);
}

__global__ __launch_bounds__(THREADS) void fused_pairs(
    const float* __restrict__ h,      // [100000,64]
    const float* __restrict__ feat,   // [500000,256]
    const float* __restrict__ u1,     // [100000,128]
    const float* __restrict__ phi_w,  // [64,256]
    const float* __restrict__ phi_b,  // [64]
    const float* __restrict__ w1,     // [64,256]
    const float* __restrict__ b1,     // [64]
    const float* __restrict__ w2,     // [64,64]
    const float* __restrict__ b2,     // [64]
    const float* __restrict__ w3,     // [64]
    const float* __restrict__ b3,     // [1]
    const float* __restrict__ vw,     // [256]
    const int* __restrict__ c_adj,    // [2,500000]
    const int* __restrict__ jv,       // [1]
    float* __restrict__ out,          // out[0]=value, out[1..NPAIRS]=a_values
    double* __restrict__ vacc)        // workspace accumulator
{
  // Per-wave B-operand staging buffer, K-pair interleaved:
  // element (k, pair) lives at ustage[wid][k>>1][(pair<<1) | (k&1)]
  __shared__ __align__(16) float ustage[WPB][64][32];  // 32 KB

  const int tid = threadIdx.x;
  const int wid = tid >> 5;
  const int lane = tid & 31;
  const int lp = lane & 15;  // pair-in-tile (B/C column)
  const int lh = lane >> 4;  // half-wave: +2 on k, +8 on channel
  const int k0b = 2 * lh;

  const long wtile = (long)blockIdx.x * WPB + wid;
  const long pbase = wtile * 16;
  if (pbase >= NPAIRS) return;  // whole wave exits; no barriers used anywhere

  const int j = jv[0];
  int nld = 0, eld = 0;
  if (lane < 16) {
    nld = c_adj[pbase + lane];
    eld = c_adj[NPAIRS + pbase + lane] - j;
  }
  const int nid = __shfl(nld, lp, 32);
  const int eid = __shfl(eld, lp, 32);

  const float* frow = feat + (long)eid * FEAT;
  const float* hrow = h + (long)nid * EMB;
  const float* urow = u1 + (long)nid * U1DIM;

  const v8f vzero = {0.f, 0.f, 0.f, 0.f, 0.f, 0.f, 0.f, 0.f};

  // ---------------- Phase 1: a_e^T = phi_w @ feat_row^T ----------------
  v8f acc[4];
#pragma unroll
  for (int i = 0; i < 4; ++i) acc[i] = vzero;

#pragma unroll 4
  for (int kb = 0; kb < FEAT; kb += 4) {
    const int k0 = kb + k0b;
    v2f b = *(const v2f*)(frow + k0);
#pragma unroll
    for (int mt = 0; mt < 4; ++mt) {
      v2f a = *(const v2f*)(phi_w + (mt * 16 + lp) * FEAT + k0);
      acc[mt] = wmma4(a, b, acc[mt]);
    }
  }

  // bias, gather h, sum of squares of a_h
  float hv[4][8];
  float ss = 0.f;
  float tot = 0.f;
#pragma unroll
  for (int mt = 0; mt < 4; ++mt) {
#pragma unroll
    for (int r = 0; r < 8; ++r) {
      const int ch = mt * 16 + 8 * lh + r;
      float ae = acc[mt][r] + phi_b[ch];
      acc[mt][r] = ae;
      float hvv = hrow[ch];
      hv[mt][r] = hvv;
      float av = ae * hvv;
      ss += av * av;
    }
  }
  ss += __shfl_xor(ss, 16, 32);  // full 64-ch sumsq for pair lp
  const float inv = 1.0f / fmaxf(sqrtf(ss), 1e-12f);

  // relu(a_e) -> U[0:64], a_h -> U[64:128]; stage B-packed; fold into <U,vw>
#pragma unroll
  for (int mt = 0; mt < 4; ++mt) {
#pragma unroll
    for (int r = 0; r < 8; ++r) {
      const int ch = mt * 16 + 8 * lh + r;
      const float ae = acc[mt][r];
      const float rv = fmaxf(ae, 0.f);
      const float av = ae * hv[mt][r] * inv;
      tot += rv * vw[ch] + av * vw[EMB + ch];
      ustage[wid][ch >> 1][(lp << 1) | (ch & 1)] = rv;
      ustage[wid][32 + (ch >> 1)][(lp << 1) | (ch & 1)] = av;
    }
  }

  // ---------------- Phase 2: x1^T = relu(w1 @ U^T + b1) ----------------
  v8f acc2[4];
#pragma unroll
  for (int i = 0; i < 4; ++i) acc2[i] = vzero;

#pragma unroll 4
  for (int kb = 0; kb < 128; kb += 4) {  // k = 0..127 from LDS stage
    const int k0 = kb + k0b;
    v2f b = *(const v2f*)&ustage[wid][k0 >> 1][lp << 1];
#pragma unroll
    for (int mt = 0; mt < 4; ++mt) {
      v2f a = *(const v2f*)(w1 + (mt * 16 + lp) * FEAT + k0);
      acc2[mt] = wmma4(a, b, acc2[mt]);
    }
  }
#pragma unroll 4
  for (int kb = 128; kb < 256; kb += 4) {  // k = 128..255 from u1 gather
    const int k0 = kb + k0b;
    v2f b = *(const v2f*)(urow + (k0 - 128));
    tot += b.x * vw[k0] + b.y * vw[k0 + 1];  // u1 block of <U,vw>
#pragma unroll
    for (int mt = 0; mt < 4; ++mt) {
      v2f a = *(const v2f*)(w1 + (mt * 16 + lp) * FEAT + k0);
      acc2[mt] = wmma4(a, b, acc2[mt]);
    }
  }
  // relu + restage x1^T into B-packed layout (khalf 0..31; phase-1 data dead)
#pragma unroll
  for (int mt = 0; mt < 4; ++mt) {
#pragma unroll
    for (int r = 0; r < 8; ++r) {
      const int ch = mt * 16 + 8 * lh + r;
      const float x = fmaxf(acc2[mt][r] + b1[ch], 0.f);
      ustage[wid][ch >> 1][(lp << 1) | (ch & 1)] = x;
    }
  }

  // ---------------- Phase 3: x2^T = relu(w2 @ x1^T + b2) ----------------
  v8f acc3[4];
#pragma unroll
  for (int i = 0; i < 4; ++i) acc3[i] = vzero;

#pragma unroll 4
  for (int kb = 0; kb < 64; kb += 4) {
    const int k0 = kb + k0b;
    v2f b = *(const v2f*)&ustage[wid][k0 >> 1][lp << 1];
#pragma unroll
    for (int mt = 0; mt < 4; ++mt) {
      v2f a = *(const v2f*)(w2 + (mt * 16 + lp) * EMB + k0);
      acc3[mt] = wmma4(a, b, acc3[mt]);
    }
  }

  // ---------------- Phase 4: head a_values = x2 . w3 + b3 ----------------
  float part = 0.f;
#pragma unroll
  for (int mt = 0; mt < 4; ++mt) {
#pragma unroll
    for (int r = 0; r < 8; ++r) {
      const int ch = mt * 16 + 8 * lh + r;
      const float x = fmaxf(acc3[mt][r] + b2[ch], 0.f);
      part += x * w3[ch];
    }
  }
  part += __shfl_xor(part, 16, 32);  // combine channel halves per pair
  if (lane < 16) out[1 + pbase + lane] = part + b3[0];

  // ---------------- Phase 5: global <U,vw> accumulation ----------------
#pragma unroll
  for (int off = 16; off > 0; off >>= 1) tot += __shfl_xor(tot, off, 32);
  if (lane == 0) atomicAdd(vacc, (double)tot);
}

__global__ void finalize_value(const double* __restrict__ vacc,
                               const float* __restrict__ vb,
                               float* __restrict__ out) {
  out[0] = (float)(vacc[0] / (double)NPAIRS + (double)vb[0]);
}

extern "C" void kernel_launch(void* const* d_in, const int* in_sizes, int n_in,
                              void* d_out, int out_size, void* d_ws,
                              size_t ws_size, hipStream_t stream) {
  const float* h     = (const float*)d_in[0];
  const float* feat  = (const float*)d_in[1];
  const float* u1    = (const float*)d_in[2];
  const float* phi_w = (const float*)d_in[3];
  const float* phi_b = (const float*)d_in[4];
  const float* w1    = (const float*)d_in[5];
  const float* b1    = (const float*)d_in[6];
  const float* w2    = (const float*)d_in[7];
  const float* b2    = (const float*)d_in[8];
  const float* w3    = (const float*)d_in[9];
  const float* b3    = (const float*)d_in[10];
  const float* vw    = (const float*)d_in[11];
  const float* vb    = (const float*)d_in[12];
  const int* c_adj   = (const int*)d_in[13];
  const int* jv      = (const int*)d_in[14];

  float* out = (float*)d_out;
  double* vacc = (double*)d_ws;

  hipMemsetAsync(d_ws, 0, sizeof(double), stream);  // capture-safe

  const int nwt = (NPAIRS + 15) / 16;           // 31250 wave-tiles (exact)
  const int nblk = (nwt + WPB - 1) / WPB;       // 7813 blocks of 4 waves
  fused_pairs<<<nblk, THREADS, 0, stream>>>(h, feat, u1, phi_w, phi_b, w1, b1,
                                            w2, b2, w3, b3, vw, c_adj, jv, out,
                                            vacc);
  finalize_value<<<1, 1, 0, stream>>>(vacc, vb, out);
}